// SuperLoss_55027120996395
// MI455X (gfx1250) — compile-verified
//
#include <hip/hip_runtime.h>

// ---------------------------------------------------------------------------
// SuperLoss: mean over N of  (l - tau)*sigma + lam*log(sigma)^2,
// sigma = exp(-W0(y)), y = 0.5*max(-2/e, (l - tau)/lam), tau=log(330), lam=.25
// log(sigma)^2 == w^2 (sigma = e^-w), saving a log per element.
// ---------------------------------------------------------------------------

typedef __attribute__((ext_vector_type(2))) float v2f;
typedef __attribute__((ext_vector_type(8))) float v8f;

#define SL_TAU      5.79909265446052548f   // log(330)
#define SL_E        2.71828182845904523f
#define SL_NEG2_E  -0.73575888234288465f   // -2/e
#define SL_NITER    6                      // Halley converges to fp32 by ~4

__device__ __forceinline__ float superloss_elem(float l) {
    float beta = (l - SL_TAU) * 4.0f;                 // (l - tau) / 0.25
    float y    = 0.5f * fmaxf(SL_NEG2_E, beta);

    // Lambert W0 initial guess (matches reference)
    float p2 = fmaxf(2.0f * (SL_E * y + 1.0f), 0.0f);
    float p  = sqrtf(p2);
    float wb = -1.0f + p * (1.0f - 0.33333333333333333f * p);
    float wl = __logf(1.0f + fmaxf(y, 0.0f));         // log1p fine: selected only y>=0.5
    float w  = (y < 0.5f) ? wb : wl;

#pragma unroll
    for (int it = 0; it < SL_NITER; ++it) {
        float ew  = __expf(w);
        float f   = __fmaf_rn(w, ew, -y);             // w*e^w - y
        float wp1 = w + 1.0f;
        float den = ew * wp1 - (w + 2.0f) * f / (2.0f * wp1);
        w = w - f / den;
    }

    float sigma = __expf(-w);
    return (l - SL_TAU) * sigma + 0.25f * w * w;      // lam * log(sigma)^2 == 0.25*w^2
}

// 256-value block reduction. Wave 0 reduces via V_WMMA_F32_16X16X4_F32:
// four chained D = A*Ones + C ops sum all 256 partials (any bijection of the
// 256 values onto the A tiles is valid since B is all ones). Result valid in
// tid==0 (lanes 0 and 16 hold column 0 of rows 0-7 / 8-15 of C).
__device__ __forceinline__ float block_reduce_256(float v, float* lds) {
    const int tid = threadIdx.x;
    lds[tid] = v;
    __syncthreads();
    float total = 0.0f;
    if (tid < 32) {   // full wave 0: EXEC all-1s as WMMA requires
#if defined(__gfx1250__) && __has_builtin(__builtin_amdgcn_wmma_f32_16x16x4_f32)
        v2f ones; ones.x = 1.0f; ones.y = 1.0f;       // 4x16 ones matrix
        v8f c = {};
#pragma unroll
        for (int i = 0; i < 4; ++i) {
            v2f a;                                    // 16x4 tile = 64 partials
            a.x = lds[i * 64 + tid];
            a.y = lds[i * 64 + 32 + tid];
            c = __builtin_amdgcn_wmma_f32_16x16x4_f32(
                    false, a, false, ones, (short)0, c, false, false);
        }
        float t = c[0] + c[1] + c[2] + c[3] + c[4] + c[5] + c[6] + c[7];
        lds[tid] = t;                                 // same-wave DS, in order
        total = lds[0] + lds[16];
#else
        float t = 0.0f;
#pragma unroll
        for (int i = 0; i < 8; ++i) t += lds[tid + i * 32];
#pragma unroll
        for (int off = 16; off > 0; off >>= 1) t += __shfl_down(t, off, 32);
        total = t;                                    // valid in lane 0
#endif
    }
    return total;                                     // use only at tid==0
}

__global__ __launch_bounds__(256)
void superloss_partial_kernel(const float* __restrict__ x,
                              float* __restrict__ partial, int n) {
    __shared__ float lds[256];
    const int tid  = threadIdx.x;
    const int gtid = blockIdx.x * 256 + tid;
    const int T    = gridDim.x * 256;

    const int nvec = n >> 2;
    const float4* __restrict__ x4 = (const float4*)x;

    float acc = 0.0f;
    for (int i = gtid; i < nvec; i += T) {
        if (i + T < nvec) __builtin_prefetch(x4 + i + T, 0, 0);  // global_prefetch_b8
        float4 v = x4[i];
        acc += superloss_elem(v.x);
        acc += superloss_elem(v.y);
        acc += superloss_elem(v.z);
        acc += superloss_elem(v.w);
    }
    for (int k = (nvec << 2) + gtid; k < n; k += T)   // scalar tail (n % 4)
        acc += superloss_elem(x[k]);

    float total = block_reduce_256(acc, lds);
    if (tid == 0) partial[blockIdx.x] = total;
}

__global__ __launch_bounds__(256)
void superloss_final_kernel(const float* __restrict__ partial, int nblk,
                            float* __restrict__ out, float inv_n) {
    __shared__ float lds[256];
    const int tid = threadIdx.x;
    float acc = 0.0f;
    for (int i = tid; i < nblk; i += 256) acc += partial[i];  // fixed order
    float total = block_reduce_256(acc, lds);
    if (tid == 0) out[0] = total * inv_n;             // 1/2^25 exact
}

extern "C" void kernel_launch(void* const* d_in, const int* in_sizes, int n_in,
                              void* d_out, int out_size, void* d_ws, size_t ws_size,
                              hipStream_t stream) {
    const float* x = (const float*)d_in[0];
    const int n = in_sizes[0];

    const int NBLK = 2048;                 // 512K threads; 16 float4 per thread at N=2^25
    float* partial = (float*)d_ws;         // 8 KB of scratch

    superloss_partial_kernel<<<NBLK, 256, 0, stream>>>(x, partial, n);
    superloss_final_kernel<<<1, 256, 0, stream>>>(partial, NBLK, (float*)d_out,
                                                  1.0f / (float)n);
}